// SSIMLoss_85925115724301
// MI455X (gfx1250) — compile-verified
//
#include <hip/hip_runtime.h>

// SSIM loss for (32,3,512,512) fp32 on gfx1250.
//
// Roofline: 201 MB of input -> ~9us floor at 23.3 TB/s. Separable 11-tap
// Gaussian x5 quantities = ~11 GFLOP fp32, which we route through
// V_WMMA_F32_16X16X4_F32 (fp32 matrix pipe) instead of the VALU:
// each 16x16 output tile = 105 WMMAs (2 banded-Toeplitz matmul passes,
// K=28, for each of x, y, x^2, y^2, xy).

typedef __attribute__((ext_vector_type(2))) float v2f;
typedef __attribute__((ext_vector_type(8))) float v8f;

#define TILE 16
#define KW   11
#define IN   28          // TILE + KW - 1 = 26, padded to a multiple of 4
#define NKB  (IN / 4)    // 7 K-chunks of 4
#define IMG  512
#define OUTW 502         // IMG - KW + 1

// Normalized Gaussian tap: w[t] = exp(-(t-5)^2/(2*1.5^2)) / Z, Z = 3.75923142
__device__ __forceinline__ float wfilt(int t) {
  float ft = (float)(t - 5);
  float v  = __expf(-ft * ft * (1.0f / 4.5f)) * 0.26601216f;
  return (t >= 0 && t < KW) ? v : 0.0f;
}

template <int Q>
__device__ __forceinline__ float qval(float p, float d) {
  if (Q == 0) return p;
  if (Q == 1) return d;
  if (Q == 2) return p * p;
  if (Q == 3) return d * d;
  return p * d;
}

// One separable 11x11 Gaussian blur of quantity Q over the staged 28x28
// patch; returns the 16x16 valid-conv result in WMMA C-layout
// (lane = column n + half-split rows, 8 rows per lane).
template <int Q>
__device__ __forceinline__ v8f conv_tile(const float (*sP)[IN + 1],
                                         const float (*sD)[IN + 1],
                                         float (*sH)[TILE + 1],
                                         int lane,
                                         const float* wlo, const float* whi) {
  const int half = lane >> 4;   // 0: lanes 0-15, 1: lanes 16-31
  const int l15  = lane & 15;

  // ---- Horizontal pass: Dh(28x16) = In(28x28) x Toeplitz(28x16) ----
  // Done as two 16-row groups (rows >= 28 are zero and masked on store).
#pragma unroll
  for (int g = 0; g < 2; ++g) {
    v8f acc = {};
#pragma unroll
    for (int kb = 0; kb < NKB; ++kb) {
      const int koff = kb * 4 + half * 2;      // K index pair for this lane
      const int row  = g * 16 + l15;           // A-matrix row (M) for this lane
      v2f a, b;
      if (row < IN) {
        a.x = qval<Q>(sP[row][koff],     sD[row][koff]);
        a.y = qval<Q>(sP[row][koff + 1], sD[row][koff + 1]);
      } else {
        a.x = 0.0f; a.y = 0.0f;
      }
      b.x = wlo[kb];                           // T[k][n] = w[k - n]
      b.y = whi[kb];
      acc = __builtin_amdgcn_wmma_f32_16x16x4_f32(
          false, a, false, b, (short)0, acc, false, false);
    }
    // Spill C-layout result into LDS so it can be re-read in B-layout.
#pragma unroll
    for (int i = 0; i < 8; ++i) {
      const int m = g * 16 + half * 8 + i;
      if (m < IN) sH[m][l15] = acc[i];
    }
  }
  __syncthreads();

  // ---- Vertical pass: D(16x16) = Toeplitz^T(16x28) x Dh(28x16) ----
  v8f acc = {};
#pragma unroll
  for (int kb = 0; kb < NKB; ++kb) {
    const int koff = kb * 4 + half * 2;
    v2f a, b;
    a.x = wlo[kb];                             // W[m][k] = w[k - m], m = l15
    a.y = whi[kb];
    b.x = sH[koff][l15];
    b.y = sH[koff + 1][l15];
    acc = __builtin_amdgcn_wmma_f32_16x16x4_f32(
        false, a, false, b, (short)0, acc, false, false);
  }
  __syncthreads();                             // sH is reused by next quantity
  return acc;
}

__global__ __launch_bounds__(32)
void ssim_tile_kernel(const float* __restrict__ data,
                      const float* __restrict__ pred,
                      double* __restrict__ acc) {
  __shared__ float sP[IN][IN + 1];
  __shared__ float sD[IN][IN + 1];
  __shared__ float sH[IN][TILE + 1];

  const int lane = threadIdx.x;
  const int r0   = blockIdx.y * TILE;
  const int c0   = blockIdx.x * TILE;
  const size_t img = (size_t)blockIdx.z * (size_t)(IMG * IMG);

  // Stage 28x28 patch of pred & data (zero-fill outside the image; those
  // values only feed outputs masked away below, and zeros avoid NaN*0).
  for (int idx = lane; idx < IN * IN; idx += 32) {
    const int row = idx / IN, col = idx % IN;
    const int gr = r0 + row, gc = c0 + col;
    const bool inb = (gr < IMG) && (gc < IMG);
    const size_t off = img + (size_t)gr * IMG + gc;
    sP[row][col] = inb ? pred[off] : 0.0f;
    sD[row][col] = inb ? data[off] : 0.0f;
  }
  __syncthreads();

  // Per-lane banded-Toeplitz weight fragments, identical for the B-matrix of
  // the horizontal pass and the A-matrix of the vertical pass. Fully
  // unrolled -> lives in registers.
  const int half = lane >> 4;
  const int l15  = lane & 15;
  float wlo[NKB], whi[NKB];
#pragma unroll
  for (int kb = 0; kb < NKB; ++kb) {
    const int t0 = kb * 4 + half * 2 - l15;
    wlo[kb] = wfilt(t0);
    whi[kb] = wfilt(t0 + 1);
  }

  const v8f mu_x  = conv_tile<0>(sP, sD, sH, lane, wlo, whi);
  const v8f mu_y  = conv_tile<1>(sP, sD, sH, lane, wlo, whi);
  const v8f mu_xx = conv_tile<2>(sP, sD, sH, lane, wlo, whi);
  const v8f mu_yy = conv_tile<3>(sP, sD, sH, lane, wlo, whi);
  const v8f mu_xy = conv_tile<4>(sP, sD, sH, lane, wlo, whi);

  const float c1 = 1.0e-4f;  // (0.01*1.0)^2
  const float c2 = 9.0e-4f;  // (0.03*1.0)^2
  float sum = 0.0f;
#pragma unroll
  for (int i = 0; i < 8; ++i) {
    const int orow = r0 + half * 8 + i;
    const int ocol = c0 + l15;
    const float mx = mu_x[i],  my = mu_y[i];
    const float sx  = mu_xx[i] - mx * mx;
    const float sy  = mu_yy[i] - my * my;
    const float sxy = mu_xy[i] - mx * my;
    const float cs   = (2.0f * sxy + c2) / (sx + sy + c2);
    const float ssim = (2.0f * mx * my + c1) / (mx * mx + my * my + c1) * cs;
    if (orow < OUTW && ocol < OUTW) sum += ssim;
  }

  // wave32 reduction, then one f64 atomic per tile (fp32 accumulation of
  // ~24M terms would lose precision).
  for (int off = 16; off > 0; off >>= 1)
    sum += __shfl_xor(sum, off, 32);
  if (lane == 0) atomicAdd(acc, (double)sum);
}

__global__ void ssim_init_kernel(double* acc) { *acc = 0.0; }

__global__ void ssim_finalize_kernel(const double* acc, float* out) {
  const double N = 32.0 * 3.0 * (double)OUTW * (double)OUTW;
  out[0] = (float)(1.0 - (*acc) / N);
}

extern "C" void kernel_launch(void* const* d_in, const int* in_sizes, int n_in,
                              void* d_out, int out_size, void* d_ws, size_t ws_size,
                              hipStream_t stream) {
  (void)n_in; (void)out_size; (void)ws_size;
  const float* data = (const float*)d_in[0];   // setup_inputs: "data" first
  const float* pred = (const float*)d_in[1];   // then "prediction"
  float* out  = (float*)d_out;
  double* acc = (double*)d_ws;

  const int bc = in_sizes[0] / (IMG * IMG);    // 96 channel-images
  const int nt = (IMG + TILE - 1) / TILE;      // 32 tiles per axis

  ssim_init_kernel<<<1, 1, 0, stream>>>(acc);
  dim3 grid(nt, nt, bc);
  ssim_tile_kernel<<<grid, 32, 0, stream>>>(data, pred, acc);
  ssim_finalize_kernel<<<1, 1, 0, stream>>>(acc, out);
}